// MeanSimilarityLoss_8358006358072
// MI455X (gfx1250) — compile-verified
//
#include <hip/hip_runtime.h>
#include <hip/hip_bf16.h>

typedef float v2f __attribute__((ext_vector_type(2)));
typedef float v8f __attribute__((ext_vector_type(8)));

__device__ __forceinline__ float wave_red_sum(float v) {
#pragma unroll
    for (int off = 16; off > 0; off >>= 1) v += __shfl_xor(v, off, 32);
    return v;
}

// ---------------- K0: zero accumulators (must run every call) ----------------
__global__ void msl_init(int* __restrict__ counts_i, float* __restrict__ colsum,
                         float* __restrict__ colsumsq, float* __restrict__ scalars, int U_) {
    int i = blockIdx.x * blockDim.x + threadIdx.x;
    if (i < U_) counts_i[i] = 0;
    if (i < 256) { colsum[i] = 0.0f; colsumsq[i] = 0.0f; }
    if (i < 2) scalars[i] = 0.0f;
}

// ---------------- K1: histogram of segment ids ----------------
__global__ void msl_hist(const int* __restrict__ idx, int* __restrict__ counts_i, int N_) {
    int i = blockIdx.x * blockDim.x + threadIdx.x;
    if (i < N_) {
        int u = idx[i];
        if (u >= 0) atomicAdd(&counts_i[u], 1);
    }
}

// ---------------- K2: exclusive scan over U counts (single block) ----------------
__global__ void msl_scan(const int* __restrict__ counts_i, int* __restrict__ offsets,
                         int* __restrict__ cursor, float* __restrict__ scalars, int U_) {
    __shared__ int lds[256];
    const int t = threadIdx.x;
    const int per = (U_ + 255) / 256;
    int s = 0;
    for (int j = 0; j < per; ++j) {
        int u = t * per + j;
        if (u < U_) s += counts_i[u];
    }
    lds[t] = s;
    __syncthreads();
    if (t == 0) {
        int run = 0;
        for (int i = 0; i < 256; ++i) { int v = lds[i]; lds[i] = run; run += v; }
        scalars[1] = (float)run;   // number of valid rows
    }
    __syncthreads();
    int base = lds[t];
    for (int j = 0; j < per; ++j) {
        int u = t * per + j;
        if (u < U_) {
            offsets[u] = base;
            cursor[u]  = base;
            base += counts_i[u];
        }
    }
}

// ---------------- K3: scatter row indices into segment-grouped order ----------------
__global__ void msl_scatter(const int* __restrict__ idx, int* __restrict__ cursor,
                            int* __restrict__ order, int N_) {
    int i = blockIdx.x * blockDim.x + threadIdx.x;
    if (i < N_) {
        int u = idx[i];
        if (u >= 0) {
            int p = atomicAdd(&cursor[u], 1);
            order[p] = i;
        }
    }
}

// ---------------- K4: per-segment fused pass (H == 256) ----------------
// One block (8 waves) per segment. Computes, in a single read of the rows:
//   sum_u   = sum of rows,  nsum_u = sum of rows/max(||row||,1e-8)
//   segcos  = <nsum_u, sum_u>/cnt / max(||sum_u||/cnt, 1e-8)   (added to scalars[0])
//   mnorm_u = sum_u / max(||sum_u||, cnt*1e-12)                (stored, 8 MB)
//   counts  -> d_out[2+u]
__global__ void msl_segment(const float* __restrict__ E, const int* __restrict__ order,
                            const int* __restrict__ offsets, const int* __restrict__ counts_i,
                            float* __restrict__ mnorm, float* __restrict__ out_counts,
                            float* __restrict__ scalars) {
    __shared__ float lsum[256];
    __shared__ float lnsum[256];
    __shared__ float wred[16];
    __shared__ float bc[1];

    const int u    = blockIdx.x;
    const int tid  = threadIdx.x;
    const int lane = tid & 31;
    const int wave = tid >> 5;
    const int cnt  = counts_i[u];
    const int off  = offsets[u];

    lsum[tid] = 0.0f;
    lnsum[tid] = 0.0f;
    __syncthreads();

    float s[8]  = {0.f,0.f,0.f,0.f,0.f,0.f,0.f,0.f};
    float ns[8] = {0.f,0.f,0.f,0.f,0.f,0.f,0.f,0.f};

    for (int r = wave; r < cnt; r += 8) {
        const int row = order[off + r];
        const float* Erow = E + (size_t)row * 256;
        const float4 va = *(const float4*)(Erow + 4 * lane);
        const float4 vb = *(const float4*)(Erow + 128 + 4 * lane);
        float e[8] = {va.x, va.y, va.z, va.w, vb.x, vb.y, vb.z, vb.w};
        float sq = 0.0f;
#pragma unroll
        for (int j = 0; j < 8; ++j) sq += e[j] * e[j];
        sq = wave_red_sum(sq);                               // ||row||^2 on all lanes
        const float rinv = 1.0f / fmaxf(sqrtf(sq), 1e-8f);
#pragma unroll
        for (int j = 0; j < 8; ++j) { s[j] += e[j]; ns[j] += e[j] * rinv; }
    }

    // merge 8 wave-private partials via LDS float atomics (ds_add_f32)
#pragma unroll
    for (int j = 0; j < 4; ++j) {
        atomicAdd(&lsum[4 * lane + j],        s[j]);
        atomicAdd(&lsum[128 + 4 * lane + j],  s[4 + j]);
        atomicAdd(&lnsum[4 * lane + j],       ns[j]);
        atomicAdd(&lnsum[128 + 4 * lane + j], ns[4 + j]);
    }
    __syncthreads();

    const float sh = lsum[tid];
    const float nh = lnsum[tid];
    float p1 = wave_red_sum(nh * sh);   // <nsum, sum> partial
    float p2 = wave_red_sum(sh * sh);   // ||sum||^2 partial
    if (lane == 0) { wred[wave] = p1; wred[8 + wave] = p2; }
    __syncthreads();

    if (tid == 0) {
        float d1 = 0.0f, d2 = 0.0f;
#pragma unroll
        for (int w = 0; w < 8; ++w) { d1 += wred[w]; d2 += wred[8 + w]; }
        const float fc = (float)cnt;
        out_counts[u] = fc;
        float rcpm = 0.0f;
        if (cnt > 0) {
            const float norm = sqrtf(d2);
            const float nb = fmaxf(norm / fc, 1e-8f);          // max(||mean||, eps)
            const float segcos = (d1 / fc) / nb;                // sum of per-row cosines
            atomicAdd(&scalars[0], segcos);
            rcpm = 1.0f / fmaxf(norm, fc * 1e-12f);             // F.normalize eps on mean
        }
        bc[0] = rcpm;
    }
    __syncthreads();
    mnorm[(size_t)u * 256 + tid] = lsum[tid] * bc[0];
}

// ---------------- K5: WMMA column reduction of mnorm (U x 256) ----------------
// All-ones A => D[m][n] = sum_k B[k][n] : v_wmma_f32_16x16x4_f32 as a 4-row
// columnwise reducer, accumulated over 128 rows per block. Exact fp32.
// grid = (H/16 col tiles) * (U/128 row chunks), 32 threads (one wave).
__global__ void msl_colsum_wmma(const float* __restrict__ mnorm, float* __restrict__ colsum,
                                float* __restrict__ colsumsq) {
    const int lane   = threadIdx.x;           // 0..31, full EXEC for WMMA
    const int ctile  = blockIdx.x & 15;       // 16 column tiles of 16
    const int rchunk = blockIdx.x >> 4;       // 128-row chunks
    const int c      = ctile * 16 + (lane & 15);
    const int rsub   = (lane < 16) ? 0 : 2;   // B: lanes 0-15 hold K=0,1; 16-31 hold K=2,3

    v2f ones; ones.x = 1.0f; ones.y = 1.0f;
    v8f acc = {};
    v8f accsq = {};
    const float* base = mnorm + (size_t)(rchunk * 128) * 256 + c;

    for (int r = 0; r < 128; r += 4) {
        const float b0 = base[(size_t)(r + rsub) * 256];
        const float b1 = base[(size_t)(r + rsub + 1) * 256];
        v2f bv;  bv.x  = b0;      bv.y  = b1;
        v2f bsq; bsq.x = b0 * b0; bsq.y = b1 * b1;
        acc   = __builtin_amdgcn_wmma_f32_16x16x4_f32(false, ones, false, bv,
                                                      (short)0, acc,   false, false);
        accsq = __builtin_amdgcn_wmma_f32_16x16x4_f32(false, ones, false, bsq,
                                                      (short)0, accsq, false, false);
    }
    // every row of D holds the column sums; take M=0 row (VGPR0, lanes 0-15)
    if (lane < 16) {
        atomicAdd(&colsum[c],   acc[0]);
        atomicAdd(&colsumsq[c], accsq[0]);
    }
}

// ---------------- K6: final scalars ----------------
// sum(cos_sim) = ||S||^2 with S = colsum; trace = sum(colsumsq).
__global__ void msl_final(const float* __restrict__ colsum, const float* __restrict__ colsumsq,
                          const float* __restrict__ scalars, float* __restrict__ d_out, int U_) {
    const int lane = threadIdx.x;
    float s2 = 0.0f, t = 0.0f;
    for (int h = lane; h < 256; h += 32) {
        const float cs = colsum[h];
        s2 += cs * cs;
        t  += colsumsq[h];
    }
    s2 = wave_red_sum(s2);
    t  = wave_red_sum(t);
    if (lane == 0) {
        const float segcos = scalars[0];
        const float nvalid = scalars[1];
        d_out[0] = (nvalid - segcos) / (nvalid + 1e-9f);
        const double n = (double)U_;
        d_out[1] = (float)(((double)s2 - (double)t) / (n * n - n + 1e-9));
    }
}

extern "C" void kernel_launch(void* const* d_in, const int* in_sizes, int n_in,
                              void* d_out, int out_size, void* d_ws, size_t ws_size,
                              hipStream_t stream) {
    const float* E  = (const float*)d_in[0];
    const int* idx  = (const int*)d_in[1];
    const int N_    = in_sizes[1];
    const int H_    = in_sizes[0] / N_;        // == 256 (layout below assumes this)
    const int U_    = out_size - 2;            // outputs: [avg_dis, mean_dis, counts[U]]
    float* out      = (float*)d_out;
    (void)n_in; (void)ws_size; (void)H_;

    // workspace carving (~10.6 MB total)
    char* p = (char*)d_ws;
    int*   counts_i = (int*)p;   p += (size_t)U_ * 4;
    int*   offsets  = (int*)p;   p += (size_t)U_ * 4;
    int*   cursor   = (int*)p;   p += (size_t)U_ * 4;
    int*   order    = (int*)p;   p += (size_t)N_ * 4;
    float* mnorm    = (float*)p; p += (size_t)U_ * 256 * 4;
    float* colsum   = (float*)p; p += 256 * 4;
    float* colsumsq = (float*)p; p += 256 * 4;
    float* scalars  = (float*)p; p += 64;

    msl_init<<<(U_ + 255) / 256, 256, 0, stream>>>(counts_i, colsum, colsumsq, scalars, U_);
    msl_hist<<<(N_ + 255) / 256, 256, 0, stream>>>(idx, counts_i, N_);
    msl_scan<<<1, 256, 0, stream>>>(counts_i, offsets, cursor, scalars, U_);
    msl_scatter<<<(N_ + 255) / 256, 256, 0, stream>>>(idx, cursor, order, N_);
    msl_segment<<<U_, 256, 0, stream>>>(E, order, offsets, counts_i, mnorm, out + 2, scalars);
    msl_colsum_wmma<<<16 * (U_ / 128), 32, 0, stream>>>(mnorm, colsum, colsumsq);
    msl_final<<<1, 32, 0, stream>>>(colsum, colsumsq, scalars, out, U_);
}